// NNSensorResponse_31353261261376
// MI455X (gfx1250) — compile-verified
//
#include <hip/hip_runtime.h>
#include <hip/hip_bf16.h>

// -------------------------------------------------------------------------
// NNSensorResponse for MI455X (gfx1250, wave32).
//
// Phase 1 (mlp_kernel): per-electron MLP 2->128->128->64 with
//   V_WMMA_F32_16X16X4_F32 (full-f32 matrix path, faithful to reference).
//   Writes resp[N,64] = el_photons * sigmoid(mlp(xy)) to workspace.
// Phase 2 (scatter_kernel): Gaussian soft-binning is numerically local
//   (sigma=1 -> exp(-d^2/2) < 6e-32 for |d|>12), so instead of the dense
//   [N,1024] einsum we scatter each electron's 64 sensor responses into a
//   +/-12-tick window of an LDS-privatized 64x1024 waveform (256KB LDS,
//   ds_add_f32), then flush per-block partials with global f32 atomics.
// -------------------------------------------------------------------------

typedef __attribute__((ext_vector_type(2))) float v2f;
typedef __attribute__((ext_vector_type(8))) float v8f;

#define HP 130            // LDS pitch (floats) for the 128x128 h2 tile
#define TICKS 1024
#define SENSORS 64
#define INV_SQRT_2PI 0.3989422804f

__global__ void __launch_bounds__(256)
mlp_kernel(const float* __restrict__ photons,   // [N,1]
           const float* __restrict__ xy,        // [N,2]
           const float* __restrict__ W1,        // [2,128]
           const float* __restrict__ b1,        // [128]
           const float* __restrict__ W2,        // [128,128]
           const float* __restrict__ b2,        // [128]
           const float* __restrict__ W3,        // [128,64]
           const float* __restrict__ b3,        // [64]
           float* __restrict__ resp,            // [N,64] out (workspace)
           int N)
{
    __shared__ float sh_h2[128 * HP];           // ~66.6 KB

    const int tid  = threadIdx.x;
    const int wave = tid >> 5;                  // 0..7 -> row tile [wave*16, +16)
    const int lane = tid & 31;
    const int n    = lane & 15;                 // column-within-tile / row M
    const int hi   = lane >> 4;                 // half-wave selector
    const int klo  = hi * 2;                    // K lanes 0-15 -> {0,1}; 16-31 -> {2,3}

    const int rowL  = wave * 16 + n;            // local electron row 0..127
    int       grow  = blockIdx.x * 128 + rowL;  // global electron row
    int       growc = grow < N ? grow : (N - 1);

    // ---------------- Layer 1 (K=2): scalar FMAs straight into the
    // 16x4-f32 A-fragment register layout: lane holds h[rowL][kk*4+klo+i].
    const float x = xy[2 * (size_t)growc + 0];
    const float y = xy[2 * (size_t)growc + 1];
    float hreg[64];
#pragma unroll
    for (int kk = 0; kk < 32; ++kk) {
#pragma unroll
        for (int i = 0; i < 2; ++i) {
            const int k = kk * 4 + klo + i;
            float v = fmaf(x, W1[k], fmaf(y, W1[128 + k], b1[k]));
            hreg[kk * 2 + i] = v > 0.f ? v : 0.f;
        }
    }

    // ---------------- Layer 2: [16x128] @ [128x128], 8 col tiles, K in steps of 4
    const v8f vzero = {0.f, 0.f, 0.f, 0.f, 0.f, 0.f, 0.f, 0.f};
    v8f acc2[8];
#pragma unroll
    for (int ct = 0; ct < 8; ++ct) acc2[ct] = vzero;

#pragma unroll
    for (int kk = 0; kk < 32; ++kk) {
        v2f a;
        a.x = hreg[kk * 2 + 0];
        a.y = hreg[kk * 2 + 1];
        const int kb = kk * 4 + klo;
#pragma unroll
        for (int ct = 0; ct < 8; ++ct) {
            v2f b;                               // B frag: W2[kb+i][ct*16+n]
            b.x = W2[(size_t)(kb + 0) * 128 + ct * 16 + n];
            b.y = W2[(size_t)(kb + 1) * 128 + ct * 16 + n];
            acc2[ct] = __builtin_amdgcn_wmma_f32_16x16x4_f32(
                false, a, false, b, (short)0, acc2[ct], false, false);
        }
    }

    // bias + relu, stage h2 in LDS (each wave writes/reads only its own rows)
#pragma unroll
    for (int ct = 0; ct < 8; ++ct) {
        const int col = ct * 16 + n;
        const float bb = b2[col];
#pragma unroll
        for (int v = 0; v < 8; ++v) {
            const int m = v + hi * 8;            // D-frag row mapping
            float h = acc2[ct][v] + bb;
            sh_h2[(wave * 16 + m) * HP + col] = h > 0.f ? h : 0.f;
        }
    }
    __syncthreads();

    // ---------------- Layer 3: [16x128] @ [128x64], 4 col tiles
    v8f acc3[4];
#pragma unroll
    for (int ct = 0; ct < 4; ++ct) acc3[ct] = vzero;

#pragma unroll
    for (int kk = 0; kk < 32; ++kk) {
        const int kb = kk * 4 + klo;
        const float* pa = &sh_h2[rowL * HP + kb]; // 8B-aligned (HP even, kb even)
        v2f a;
        a.x = pa[0];
        a.y = pa[1];
#pragma unroll
        for (int ct = 0; ct < 4; ++ct) {
            v2f b;
            b.x = W3[(size_t)(kb + 0) * 64 + ct * 16 + n];
            b.y = W3[(size_t)(kb + 1) * 64 + ct * 16 + n];
            acc3[ct] = __builtin_amdgcn_wmma_f32_16x16x4_f32(
                false, a, false, b, (short)0, acc3[ct], false, false);
        }
    }

    // ---------------- Epilogue: bias, sigmoid, photon scale, store [N,64]
#pragma unroll
    for (int v = 0; v < 8; ++v) {
        const int m  = v + hi * 8;
        const int gr = blockIdx.x * 128 + wave * 16 + m;
        if (gr >= N) continue;
        const float ph = photons[gr];
#pragma unroll
        for (int ct = 0; ct < 4; ++ct) {
            const int col = ct * 16 + n;
            const float val = acc3[ct][v] + b3[col];
            const float sg  = 1.0f / (1.0f + __expf(-val));
            resp[(size_t)gr * 64 + col] = ph * sg;
        }
    }
}

// -------------------------------------------------------------------------
// Windowed Gaussian scatter with per-block LDS waveform privatization.
// acc layout [tick][sensor] -> 64 sensor lanes hit 64 distinct LDS banks.
// -------------------------------------------------------------------------
__global__ void __launch_bounds__(1024)
scatter_kernel(const float* __restrict__ resp,  // [N,64]
               const float* __restrict__ z,     // [N]
               float* __restrict__ out,         // [64,1024]
               int N)
{
    extern __shared__ float acc[];              // TICKS*SENSORS floats = 256 KB

    const int tid = threadIdx.x;
    for (int i = tid; i < TICKS * SENSORS; i += blockDim.x) acc[i] = 0.f;
    __syncthreads();

    const int s      = tid & 63;                // sensor
    const int g      = tid >> 6;                // electron group within block
    const int groups = blockDim.x >> 6;         // 16

    for (int e = blockIdx.x * groups + g; e < N; e += gridDim.x * groups) {
        const float ze = z[e];
        const float r  = resp[(size_t)e * 64 + s];
        const int zi = (int)ze;
        int t0 = zi - 11; if (t0 < 0) t0 = 0;
        int t1 = zi + 12; if (t1 > TICKS - 1) t1 = TICKS - 1;
        for (int t = t0; t <= t1; ++t) {
            const float d = (float)t - ze;
            const float w = __expf(-0.5f * d * d) * INV_SQRT_2PI;
            atomicAdd(&acc[t * SENSORS + s], r * w);   // ds_add_f32
        }
    }
    __syncthreads();

    // flush partials: out[s][t] += acc[t][s]
    for (int i = tid; i < TICKS * SENSORS; i += blockDim.x) {
        const int t  = i >> 6;
        const int ss = i & 63;
        const float v = acc[i];
        atomicAdd(&out[(size_t)ss * TICKS + t], v);
    }
}

__global__ void zero_out_kernel(float* __restrict__ out, int n)
{
    const int i = blockIdx.x * blockDim.x + threadIdx.x;
    if (i < n) out[i] = 0.f;
}

extern "C" void kernel_launch(void* const* d_in, const int* in_sizes, int n_in,
                              void* d_out, int out_size, void* d_ws, size_t ws_size,
                              hipStream_t stream)
{
    const float* photons = (const float*)d_in[0];   // [N,1]
    const float* xy      = (const float*)d_in[1];   // [N,2]
    const float* z       = (const float*)d_in[2];   // [N]
    const float* W1      = (const float*)d_in[3];
    const float* b1      = (const float*)d_in[4];
    const float* W2      = (const float*)d_in[5];
    const float* b2      = (const float*)d_in[6];
    const float* W3      = (const float*)d_in[7];
    const float* b3      = (const float*)d_in[8];
    float* out  = (float*)d_out;
    float* resp = (float*)d_ws;                     // [N,64] f32 scratch

    const int N = in_sizes[2];

    // Allow the 256 KB dynamic-LDS allocation (no-op if unsupported).
    (void)hipFuncSetAttribute((const void*)scatter_kernel,
                              hipFuncAttributeMaxDynamicSharedMemorySize,
                              TICKS * SENSORS * (int)sizeof(float));

    // 1) zero output accumulator (harness poisons d_out)
    zero_out_kernel<<<(out_size + 255) / 256, 256, 0, stream>>>(out, out_size);

    // 2) MLP -> resp
    const int nwg = (N + 127) / 128;
    mlp_kernel<<<nwg, 256, 0, stream>>>(photons, xy, W1, b1, W2, b2, W3, b3,
                                        resp, N);

    // 3) windowed Gaussian scatter into [64,1024] waveforms
    scatter_kernel<<<192, 1024, TICKS * SENSORS * sizeof(float), stream>>>(
        resp, z, out, N);
}